// PerspectiveRasterizer_82128364634432
// MI455X (gfx1250) — compile-verified
//
#include <hip/hip_runtime.h>
#include <stdint.h>

// ---------------------------------------------------------------------------
// PerspectiveRasterizer attribute interpolation for MI455X (gfx1250).
//
// Memory-bound gather kernel. Uses CDNA5 async global->LDS loads
// (GLOBAL_LOAD_ASYNC_TO_LDS_B128, ASYNCcnt) to double-buffer the random
// 96-byte attribute-row gathers in LDS, and NT cache hints to preserve
// L2 residency of the 76.8 MB attribute table (fits in 192 MB L2).
//
// Pipeline per tile i (256 pixels/block-tile, 8 tiles/block):
//   1. issue 6x async b128 gathers for tile i+1 into LDS buf (i+1)&1
//   2. start NT stream loads: bary(i), pix(i+2)        (latency overlapped)
//   3. s_wait_asynccnt <= 6  (in-order completion => tile i's 6 ops done)
//   4. ds_load_b128 x6, 24 FMAs, 9 coalesced NT b32 stores
// ---------------------------------------------------------------------------

namespace {
constexpr int kN      = 8;
constexpr int kH      = 512;
constexpr int kW      = 512;
constexpr int kD      = 8;
constexpr int kHW     = kH * kW;                    // 262144
constexpr int kPix    = kN * kHW;                   // 2097152
constexpr int kBlock  = 256;                        // 8 waves (wave32)
constexpr int kBlocks = 1024;
constexpr int kTiles  = kPix / (kBlock * kBlocks);  // 8, exact
constexpr int kRowB   = 96;                         // 3 verts * 8 ch * 4 B
constexpr int kSlot   = 112;                        // 96 + 16 pad (bank spread)
constexpr int kBufB   = kBlock * kSlot;             // 28672 B per stage
} // namespace

// Issue 6 async B128 copies: one 96-byte attribute row -> this lane's LDS slot.
// INST_OFFSET is added to both the global and the LDS address (ISA 08 §4.4),
// so one base VGPR pair + one LDS-offset VGPR serves all six transfers.
__device__ __forceinline__ void issue_gather6(const char* g, uint32_t l) {
  asm volatile("global_load_async_to_lds_b128 %0, %1, off"           :: "v"(l), "v"(g) : "memory");
  asm volatile("global_load_async_to_lds_b128 %0, %1, off offset:16" :: "v"(l), "v"(g) : "memory");
  asm volatile("global_load_async_to_lds_b128 %0, %1, off offset:32" :: "v"(l), "v"(g) : "memory");
  asm volatile("global_load_async_to_lds_b128 %0, %1, off offset:48" :: "v"(l), "v"(g) : "memory");
  asm volatile("global_load_async_to_lds_b128 %0, %1, off offset:64" :: "v"(l), "v"(g) : "memory");
  asm volatile("global_load_async_to_lds_b128 %0, %1, off offset:80" :: "v"(l), "v"(g) : "memory");
}

__device__ __forceinline__ void wait_async_le6() {
  asm volatile("s_wait_asynccnt 0x6" ::: "memory");
}
__device__ __forceinline__ void wait_async_le0() {
  asm volatile("s_wait_asynccnt 0x0" ::: "memory");
}

extern "C" __global__ __launch_bounds__(kBlock)
void rasterize_interp_kernel(const int*   __restrict__ pix,
                             const float* __restrict__ bary,
                             const float* __restrict__ attr,
                             float*       __restrict__ out) {
  __shared__ __align__(16) char smem[2 * kBufB];

  const int      t      = threadIdx.x;
  const uint32_t chunk0 = (uint32_t)blockIdx.x * (kTiles * kBlock);
  // Low 32 bits of a generic LDS pointer are the LDS byte offset.
  const uint32_t lslot  = (uint32_t)(uintptr_t)(&smem[0]) + (uint32_t)(t * kSlot);
  const char*    attrB  = (const char*)attr;

  auto issue_tile = [&](int pv, int buf) {
    const int idx = pv < 0 ? 0 : pv;               // clamp invalid faces to 0
    issue_gather6(attrB + (size_t)idx * kRowB, lslot + (uint32_t)(buf * kBufB));
  };

  // --- prologue: both pix tiles in flight before the first gather issue ---
  int pv_cur = __builtin_nontemporal_load(&pix[chunk0 + 0 * kBlock + t]);
  int pv_nxt = __builtin_nontemporal_load(&pix[chunk0 + 1 * kBlock + t]);
  issue_tile(pv_cur, /*buf=*/0);

#pragma unroll
  for (int i = 0; i < kTiles; ++i) {
    const uint32_t p = chunk0 + (uint32_t)(i * kBlock + t);

    // ---- stage 1: issue gathers for tile i+1 (longest-latency first) ----
    if (i + 1 < kTiles) issue_tile(pv_nxt, (i + 1) & 1);

    // ---- stage 2: start stream loads BEFORE the async fence so their ----
    // ---- latency overlaps the wait (asm clobber blocks hoisting).     ----
    const float* bp = bary + (size_t)p * 3;
    const float b0 = __builtin_nontemporal_load(bp + 0);
    const float b1 = __builtin_nontemporal_load(bp + 1);
    const float b2 = __builtin_nontemporal_load(bp + 2);
    int pv_n2 = 0;
    if (i + 2 < kTiles)
      pv_n2 = __builtin_nontemporal_load(&pix[chunk0 + (i + 2) * kBlock + t]);

    // ---- stage 3: tile i's 6 async ops are the oldest outstanding ----
    if (i + 1 < kTiles) wait_async_le6(); else wait_async_le0();

    // ---- stage 4: LDS readback, interpolate, coalesced NT stores ----
    const float4* slot = (const float4*)(smem + (i & 1) * kBufB + t * kSlot);
    const float4 u0 = slot[0], u1 = slot[1];   // vertex 0, channels 0..7
    const float4 u2 = slot[2], u3 = slot[3];   // vertex 1
    const float4 u4 = slot[4], u5 = slot[5];   // vertex 2

    const float v0[kD] = {u0.x, u0.y, u0.z, u0.w, u1.x, u1.y, u1.z, u1.w};
    const float v1[kD] = {u2.x, u2.y, u2.z, u2.w, u3.x, u3.y, u3.z, u3.w};
    const float v2[kD] = {u4.x, u4.y, u4.z, u4.w, u5.x, u5.y, u5.z, u5.w};

    const bool     valid = (pv_cur >= 0);
    const uint32_t n     = p >> 18;                // p / (H*W)
    const uint32_t hw    = p & (kHW - 1);
    float* ob = out + ((size_t)n * (kD + 1)) * kHW + hw;

#pragma unroll
    for (int d = 0; d < kD; ++d) {
      const float r = fmaf(b2, v2[d], fmaf(b1, v1[d], b0 * v0[d]));
      __builtin_nontemporal_store(valid ? r : 0.0f, ob + (size_t)d * kHW);
    }
    __builtin_nontemporal_store(valid ? 1.0f : 0.0f, ob + (size_t)kD * kHW);

    pv_cur = pv_nxt;
    pv_nxt = pv_n2;
  }
}

extern "C" void kernel_launch(void* const* d_in, const int* in_sizes, int n_in,
                              void* d_out, int out_size, void* d_ws, size_t ws_size,
                              hipStream_t stream) {
  (void)in_sizes; (void)n_in; (void)d_ws; (void)ws_size; (void)out_size;
  const int*   pix  = (const int*)d_in[0];    // pix_to_face (harness: int32)
  const float* bary = (const float*)d_in[1];  // [N,H,W,1,3]
  const float* attr = (const float*)d_in[2];  // [N*NF,3,8]
  float*       out  = (float*)d_out;          // [N,9,H,W]
  hipLaunchKernelGGL(rasterize_interp_kernel, dim3(kBlocks), dim3(kBlock), 0,
                     stream, pix, bary, attr, out);
}